// MultiboxLoss_22969485099928
// MI455X (gfx1250) — compile-verified
//
#include <hip/hip_runtime.h>
#include <stdint.h>

// Problem constants (from reference): N=64, P=8732, C=81, ratio=3
#define NB      64
#define PP      8732
#define CC      81
#define TOTAL   (NB * PP)           // 558848
#define TILE_P  128
#define BLK_A   128
#define NBLK_A  (TOTAL / TILE_P)    // 4366 (exact)
#define NEG_RATIO 3

typedef unsigned int v4u __attribute__((ext_vector_type(4)));
typedef int          v8i __attribute__((ext_vector_type(8)));
typedef int          v4i __attribute__((ext_vector_type(4)));

__device__ __forceinline__ float wave_red_f(float v) {
#pragma unroll
  for (int o = 16; o > 0; o >>= 1) v += __shfl_down(v, o, 32);
  return v;
}
__device__ __forceinline__ int wave_red_i(int v) {
#pragma unroll
  for (int o = 16; o > 0; o >>= 1) v += __shfl_down(v, o, 32);
  return v;
}

// -------- Kernel 1: TDM-staged softmax/CE/score + loc smooth-L1 --------
__global__ void __launch_bounds__(BLK_A)
mb_main(const float* __restrict__ conf, const float* __restrict__ locp,
        const int* __restrict__ lab, const float* __restrict__ loco,
        float* __restrict__ ce, float* __restrict__ score,
        float* __restrict__ locPart, int* __restrict__ posPart)
{
  __shared__ float tile[TILE_P * CC];   // 41472 B — one 128x81 f32 tile
  __shared__ float redf[BLK_A / 32];
  __shared__ int   redi[BLK_A / 32];

  const int t = threadIdx.x;
  const int b = blockIdx.x;
  const long long r0 = (long long)b * TILE_P;

  // ---- Tensor Data Mover: DMA tile (TILE_P rows x 81 f32) global -> LDS ----
  if (t < 32) {   // one wave issues the TDM op (EXEC ignored, once per wave)
    unsigned long long ga = (unsigned long long)(uintptr_t)conf +
                            (unsigned long long)r0 * (CC * 4ull);
    unsigned int ga_lo = (unsigned int)__builtin_amdgcn_readfirstlane((int)(ga & 0xFFFFFFFFull));
    unsigned int ga_hi = (unsigned int)__builtin_amdgcn_readfirstlane((int)((ga >> 32) & 0x01FFFFFFull));
    unsigned int lds_base = (unsigned int)(uintptr_t)(void*)&tile[0];  // low 32 bits = LDS byte addr

    // D# group 0: [1:0]=count=1 | lds_addr | global_addr[56:0] | type=2
    v4u g0 = { 1u,
               lds_base,
               ga_lo,
               ga_hi | (2u << 30) };

    // D# group 1: wg_mask=0, data_size=2(4B), dims/strides per bit layout
    v8i g1 = { (int)0x00020000,                                  // data_size=4B
               (int)((CC & 0xFFFF) << 16),                       // tensor_dim0 lo16
               (int)(((TOTAL & 0xFFFF) << 16) | ((CC >> 16) & 0xFFFF)),   // dim0 hi | dim1 lo
               (int)(((TOTAL >> 16) & 0xFFFF) | ((CC & 0xFFFF) << 16)),   // dim1 hi | tile_dim0
               (int)TILE_P,                                      // tile_dim1 | tile_dim2=0
               (int)CC,                                          // tensor_dim0_stride lo32
               0, 0 };                                           // stride hi / dim1_stride
    v4i gz4 = { 0, 0, 0, 0 };                                    // 2-D tensor: groups 2/3 unused
    v8i gz8 = { 0, 0, 0, 0, 0, 0, 0, 0 };

    __builtin_amdgcn_tensor_load_to_lds(g0, g1, gz4, gz4, gz8, 0);
    __builtin_amdgcn_s_wait_tensorcnt(0);
  }
  __syncthreads();

  // ---- per-prior reduction over 81 classes (bank-conflict-free: 81 ≡ 17 mod 64) ----
  const float* row = &tile[t * CC];
  float m = -3.4028235e38f;
#pragma unroll 9
  for (int c = 0; c < CC; ++c) m = fmaxf(m, row[c]);
  float s = 0.f;
#pragma unroll 9
  for (int c = 0; c < CC; ++c) s += __expf(row[c] - m);
  float sc = 0.f;
#pragma unroll 9
  for (int c = 0; c < CC; ++c) sc += __expf(row[c] - m) / s;  // softmax row-sum (score)

  const long long gp = r0 + t;
  const int  L   = lab[gp];
  const bool pos = (L > 0);
  if (pos) sc = 0.f;                                  // positive rows zeroed pre-sum
  const float cev = (m + __logf(s)) - row[L];         // logsumexp - logit[label]
  ce[gp]    = cev;
  score[gp] = sc;

  // ---- localization smooth-L1 (positives only) ----
  float ll = 0.f;
  if (pos) {
#pragma unroll
    for (int k = 0; k < 4; ++k) {
      float d  = locp[gp * 4 + k] - loco[gp * 4 + k];
      float ad = fabsf(d);
      ll += (ad < 1.f) ? 0.5f * d * d : (ad - 0.5f);
    }
  }

  // ---- deterministic block reduction -> per-block partials ----
  float wl = wave_red_f(ll);
  int   wp = wave_red_i(pos ? 1 : 0);
  const int wid = t >> 5, lid = t & 31;
  if (lid == 0) { redf[wid] = wl; redi[wid] = wp; }
  __syncthreads();
  if (t == 0) {
    float a = 0.f; int p = 0;
#pragma unroll
    for (int w = 0; w < BLK_A / 32; ++w) { a += redf[w]; p += redi[w]; }
    locPart[b] = a;
    posPart[b] = p;
  }
}

// -------- Kernel 2: global positive count --------
__global__ void __launch_bounds__(256)
mb_count(const int* __restrict__ posPart, int* __restrict__ numPos)
{
  __shared__ int red[256];
  const int t = threadIdx.x;
  int a = 0;
  for (int i = t; i < NBLK_A; i += 256) a += posPart[i];
  red[t] = a;
  __syncthreads();
#pragma unroll
  for (int o = 128; o > 0; o >>= 1) {
    if (t < o) red[t] += red[t + o];
    __syncthreads();
  }
  if (t == 0) *numPos = red[0];
}

// -------- Kernel 3: per-row hard-negative selection + CE sum --------
__global__ void __launch_bounds__(256)
mb_select(const float* __restrict__ score, const float* __restrict__ ce,
          const int* __restrict__ lab, const int* __restrict__ numPos,
          float* __restrict__ confPart)
{
  __shared__ float ssc[PP];     // 34928 B: row of scores resident in LDS
  __shared__ float red[256];
  const int n = blockIdx.x, t = threadIdx.x;
  const size_t base = (size_t)n * PP;

  for (int p = t; p < PP; p += 256) ssc[p] = score[base + p];
  __syncthreads();

  const int K = NEG_RATIO * (*numPos);
  float acc = 0.f;
  if (K >= PP) {
    // num_neg >= P: every prior selected (typical SSD case, ~2% positives)
    for (int p = t; p < PP; p += 256) acc += ce[base + p];
  } else {
    // exact stable-descending rank via pairwise counting in LDS
    for (int p = t; p < PP; p += 256) {
      const float sp = ssc[p];
      int rank = 0;
      for (int q = 0; q < PP; ++q) {
        const float sq = ssc[q];
        rank += (sq > sp || (sq == sp && q < p)) ? 1 : 0;
      }
      if (lab[base + p] > 0 || rank < K) acc += ce[base + p];
    }
  }
  red[t] = acc;
  __syncthreads();
#pragma unroll
  for (int o = 128; o > 0; o >>= 1) {
    if (t < o) red[t] += red[t + o];
    __syncthreads();
  }
  if (t == 0) confPart[n] = red[0];
}

// -------- Kernel 4: final fixed-order reduction --------
__global__ void __launch_bounds__(256)
mb_final(const float* __restrict__ locPart, const float* __restrict__ confPart,
         const int* __restrict__ numPos, float* __restrict__ out)
{
  __shared__ float red[256];
  const int t = threadIdx.x;
  float a = 0.f;
  for (int i = t; i < NBLK_A; i += 256) a += locPart[i];
  for (int i = t; i < NB; i += 256) a += confPart[i];
  red[t] = a;
  __syncthreads();
#pragma unroll
  for (int o = 128; o > 0; o >>= 1) {
    if (t < o) red[t] += red[t + o];
    __syncthreads();
  }
  if (t == 0) out[0] = red[0] / (float)(*numPos);
}

extern "C" void kernel_launch(void* const* d_in, const int* in_sizes, int n_in,
                              void* d_out, int out_size, void* d_ws, size_t ws_size,
                              hipStream_t stream)
{
  (void)in_sizes; (void)n_in; (void)out_size; (void)ws_size;
  const float* conf = (const float*)d_in[0];
  const float* locp = (const float*)d_in[1];
  const int*   lab  = (const int*)d_in[2];
  const float* loco = (const float*)d_in[3];

  // workspace layout (~4.5 MB)
  float* ce       = (float*)d_ws;
  float* score    = ce + TOTAL;
  float* locPart  = score + TOTAL;
  int*   posPart  = (int*)(locPart + NBLK_A);
  int*   numPos   = posPart + NBLK_A;
  float* confPart = (float*)(numPos + 1);

  mb_main  <<<NBLK_A, BLK_A, 0, stream>>>(conf, locp, lab, loco, ce, score, locPart, posPart);
  mb_count <<<1, 256, 0, stream>>>(posPart, numPos);
  mb_select<<<NB, 256, 0, stream>>>(score, ce, lab, numPos, confPart);
  mb_final <<<1, 256, 0, stream>>>(locPart, confPart, numPos, (float*)d_out);
}